// SE3MultiSelfAttention_10282151706962
// MI455X (gfx1250) — compile-verified
//
#include <hip/hip_runtime.h>
#include <math.h>

// ---------------- problem constants (match reference) ----------------
#define B_    4
#define NH    4
#define C_    32          // concatenated channel dim: 8 (l=0) + 24 (l=1)
#define N_    512         // 8*8*8 voxels
#define QD_   8           // query dim per degree
#define MAXR  256         // upper bound on #radius bins (actual ~100, runtime NR)
#define SCALE_ 11.313708498984761f   // sqrt(128)

typedef __attribute__((ext_vector_type(2))) float v2f;
typedef __attribute__((ext_vector_type(8))) float v8f;

// f32 WMMA: D(16x16) = A(16x4) * B(4x16) + C.  A/B neg illegal for f32 -> pass false.
__device__ __forceinline__ v8f wmma4(v2f a, v2f b, v8f c) {
  return __builtin_amdgcn_wmma_f32_16x16x4_f32(false, a, false, b, (short)0, c,
                                               false, false);
}

// ---------------------------------------------------------------------
// K1: QKV projection.  e[t,b,d,i,n] = sum_j emb_l[t,i,j] * x_l[b,d,j,n]
// scattered into planar QKV[t][b][h][c][n] with i=(h*8+q), c = off_l + d*8 + q.
// ---------------------------------------------------------------------
__global__ void qkv_proj(const float* __restrict__ x0, const float* __restrict__ x1,
                         const float* __restrict__ emb0, const float* __restrict__ emb1,
                         float* __restrict__ QKVr, float* __restrict__ QKVi) {
  int gid = blockIdx.x * blockDim.x + threadIdx.x;   // 786432 threads
  int n = gid & (N_ - 1); int u = gid >> 9;
  int i = u & 31;          u >>= 5;
  int dl = u & 3;          u >>= 2;                   // 0:l=0 ; 1..3: l=1,d=dl-1
  int b = u & 3;           int t = u >> 2;            // t in 0..2
  const float* xb; const float* em; int l, d;
  if (dl == 0) { l = 0; d = 0;
    xb = x0 + 2 * ((size_t)b * 1 * C_) * N_;
    em = emb0 + 2 * (size_t)t * C_ * C_;
  } else { l = 1; d = dl - 1;
    xb = x1 + 2 * (((size_t)b * 3 + d) * C_) * N_;
    em = emb1 + 2 * (size_t)t * C_ * C_;
  }
  float ar = 0.f, ai = 0.f;
  for (int j = 0; j < C_; ++j) {
    float er = em[2 * (i * C_ + j)],     ei = em[2 * (i * C_ + j) + 1];
    float xr = xb[2 * (j * N_ + n)],     xi = xb[2 * (j * N_ + n) + 1];
    ar += er * xr - ei * xi;
    ai += er * xi + ei * xr;
  }
  int h = i >> 3, q = i & 7;
  int c = (l == 0) ? q : 8 + d * QD_ + q;
  size_t o = ((size_t)(t * 16 + b * NH + h) * C_ + c) * N_ + n;
  QKVr[o] = ar; QKVi[o] = ai;
}

// ---------------------------------------------------------------------
// K2: attention scores A = Q^H K  (complex, un-scaled) via f32 WMMA.
// Ar = QrKr + QiKi ; Ai = QrKi - QiKr (conj sign via operand negate).
// Each wave computes a 16x32 strip (two m-tiles) re-using the Q operands.
// 8 waves/block; 8192 waves total, exact coverage (EXEC all-1 at WMMA).
// ---------------------------------------------------------------------
__global__ void attn_scores(const float* __restrict__ QKVr, const float* __restrict__ QKVi,
                            float* __restrict__ Ar, float* __restrict__ Ai) {
  int lane = threadIdx.x & 31;
  int wave = threadIdx.x >> 5;
  int tile = blockIdx.x * 8 + wave;           // 8192 = 16 bh * 32 nt * 16 mpair
  int bh = tile >> 9;
  int rem = tile & 511;
  int n0 = (rem >> 4) << 4;                    // query-row tile (16)
  int m0 = (rem & 15) << 5;                    // key-col strip (32)
  const float* Qr = QKVr + (size_t)(0 * 16 + bh) * C_ * N_;
  const float* Qi = QKVi + (size_t)(0 * 16 + bh) * C_ * N_;
  const float* Kr = QKVr + (size_t)(1 * 16 + bh) * C_ * N_;
  const float* Ki = QKVi + (size_t)(1 * 16 + bh) * C_ * N_;
  int row = lane & 15;
  int kh = (lane >> 4) << 1;                   // K-split: 0 | 2 per lane half
  v8f accR0 = {}, accI0 = {}, accR1 = {}, accI1 = {};
  #pragma unroll
  for (int c0 = 0; c0 < C_; c0 += 4) {
    v2f aqr, aqi, nqi, bkr0, bki0, bkr1, bki1;
    aqr.x  = Qr[(c0 + kh)     * N_ + n0 + row];
    aqr.y  = Qr[(c0 + kh + 1) * N_ + n0 + row];
    aqi.x  = Qi[(c0 + kh)     * N_ + n0 + row];
    aqi.y  = Qi[(c0 + kh + 1) * N_ + n0 + row];
    bkr0.x = Kr[(c0 + kh)     * N_ + m0 + row];
    bkr0.y = Kr[(c0 + kh + 1) * N_ + m0 + row];
    bki0.x = Ki[(c0 + kh)     * N_ + m0 + row];
    bki0.y = Ki[(c0 + kh + 1) * N_ + m0 + row];
    bkr1.x = Kr[(c0 + kh)     * N_ + m0 + 16 + row];
    bkr1.y = Kr[(c0 + kh + 1) * N_ + m0 + 16 + row];
    bki1.x = Ki[(c0 + kh)     * N_ + m0 + 16 + row];
    bki1.y = Ki[(c0 + kh + 1) * N_ + m0 + 16 + row];
    nqi.x = -aqi.x; nqi.y = -aqi.y;            // conj sign (f32 WMMA has no A-neg)
    accR0 = wmma4(aqr, bkr0, accR0);
    accR0 = wmma4(aqi, bki0, accR0);
    accI0 = wmma4(aqr, bki0, accI0);
    accI0 = wmma4(nqi, bkr0, accI0);
    accR1 = wmma4(aqr, bkr1, accR1);
    accR1 = wmma4(aqi, bki1, accR1);
    accI1 = wmma4(aqr, bki1, accI1);
    accI1 = wmma4(nqi, bkr1, accI1);
  }
  int colw = lane & 15;
  int moff = (lane >> 4) << 3;                 // C/D: M = v (+8 for hi lanes)
  #pragma unroll
  for (int v = 0; v < 8; ++v) {
    size_t o = ((size_t)bh * N_ + n0 + v + moff) * N_ + m0 + colw;
    Ar[o]      = accR0[v]; Ai[o]      = accI0[v];
    Ar[o + 16] = accR1[v]; Ai[o + 16] = accI1[v];
  }
}

// ---------------------------------------------------------------------
// K3: per attention row (b,h,n): add positional bias, /SCALE, abs, softmax.
// Bias factored via LDS radius table: S[part][r] = sum_q conj(Q)*pw0[...,r].
// ---------------------------------------------------------------------
__global__ void bias_softmax(const float* __restrict__ QKVr, const float* __restrict__ QKVi,
                             const float* __restrict__ pw0, const float* __restrict__ pw1,
                             const float* __restrict__ bas0, const float* __restrict__ bas1,
                             const int* __restrict__ radii,
                             float* __restrict__ Ar, const float* __restrict__ Ai, int NR) {
  __shared__ float S[4][MAXR][2];
  __shared__ float rowbuf[N_];
  __shared__ float red[256];
  int tid = threadIdx.x;
  int idx = blockIdx.x;                        // flat (bh, n)
  int n  = idx & (N_ - 1);
  int bh = idx >> 9;
  int h  = bh & 3;
  // stage 1: radius tables (part 0 -> l=0; parts 1..3 -> l=1, d=part-1)
  for (int t = tid; t < 4 * NR; t += 256) {
    int part = t / NR; int r = t - part * NR;
    const float* pw = (part == 0) ? pw0 : pw1;
    int cbase = (part == 0) ? 0 : 8 + (part - 1) * QD_;
    float sr = 0.f, si = 0.f;
    for (int q = 0; q < QD_; ++q) {
      size_t qo = ((size_t)(0 * 16 + bh) * C_ + cbase + q) * N_ + n;
      float qr = QKVr[qo], qi = QKVi[qo];
      size_t po = 2 * ((size_t)((0 * NH + h) * QD_ + q) * NR + r);
      float pr = pw[po], pi = pw[po + 1];
      sr += qr * pr + qi * pi;                 // conj(Q) * pw
      si += qr * pi - qi * pr;
    }
    S[part][r][0] = sr; S[part][r][1] = si;
  }
  __syncthreads();
  // stage 2: bias + scale + abs
  const float inv = 1.0f / SCALE_;
  for (int m = tid; m < N_; m += 256) {
    int r = radii[n * N_ + m];
    float br, bi;
    { float b0r = bas0[2 * ((size_t)n * N_ + m)];
      float b0i = bas0[2 * ((size_t)n * N_ + m) + 1];
      br = S[0][r][0] * b0r - S[0][r][1] * b0i;
      bi = S[0][r][0] * b0i + S[0][r][1] * b0r; }
    #pragma unroll
    for (int d = 0; d < 3; ++d) {
      float b1r = bas1[2 * (((size_t)n * 3 + d) * N_ + m)];
      float b1i = bas1[2 * (((size_t)n * 3 + d) * N_ + m) + 1];
      br += S[1 + d][r][0] * b1r - S[1 + d][r][1] * b1i;
      bi += S[1 + d][r][0] * b1i + S[1 + d][r][1] * b1r;
    }
    size_t ao = ((size_t)bh * N_ + n) * N_ + m;
    float ar = (Ar[ao] + br) * inv;
    float ai = (Ai[ao] + bi) * inv;
    rowbuf[m] = sqrtf(ar * ar + ai * ai);
  }
  __syncthreads();
  // softmax over the 512-wide row (block owns it entirely)
  float lm = -1e30f;
  for (int m = tid; m < N_; m += 256) lm = fmaxf(lm, rowbuf[m]);
  red[tid] = lm; __syncthreads();
  for (int s = 128; s > 0; s >>= 1) { if (tid < s) red[tid] = fmaxf(red[tid], red[tid + s]); __syncthreads(); }
  float mx = red[0]; __syncthreads();
  float ls = 0.f;
  for (int m = tid; m < N_; m += 256) { float e = __expf(rowbuf[m] - mx); rowbuf[m] = e; ls += e; }
  red[tid] = ls; __syncthreads();
  for (int s = 128; s > 0; s >>= 1) { if (tid < s) red[tid] += red[tid + s]; __syncthreads(); }
  float rinv = 1.0f / red[0];
  __syncthreads();
  for (int m = tid; m < N_; m += 256)
    Ar[((size_t)bh * N_ + n) * N_ + m] = rowbuf[m] * rinv;   // overwrite with real softmax
}

// ---------------------------------------------------------------------
// K4: Vo = V @ A^T  (V complex, A real post-softmax) via f32 WMMA, K=512.
// Each wave computes the full 32-row c-extent (both c-tiles), sharing the
// attention B-operand; operand pairs are contiguous -> b64 vector loads.
// ---------------------------------------------------------------------
__global__ void vo_gemm(const float* __restrict__ QKVr, const float* __restrict__ QKVi,
                        const float* __restrict__ A,
                        float* __restrict__ Vor, float* __restrict__ Voi) {
  int lane = threadIdx.x & 31;
  int wave = threadIdx.x >> 5;
  int tile = blockIdx.x * 8 + wave;            // 512 = 16 bh * 32 n-tiles
  int bh = tile >> 5;
  int n0 = (tile & 31) << 4;
  const float* Vr = QKVr + (size_t)(2 * 16 + bh) * C_ * N_;
  const float* Vi = QKVi + (size_t)(2 * 16 + bh) * C_ * N_;
  const float* Ab = A + (size_t)bh * N_ * N_;
  int row = lane & 15;
  int kh = (lane >> 4) << 1;
  const v2f* Vr0 = (const v2f*)(Vr + (size_t)(row)      * N_);
  const v2f* Vr1 = (const v2f*)(Vr + (size_t)(16 + row) * N_);
  const v2f* Vi0 = (const v2f*)(Vi + (size_t)(row)      * N_);
  const v2f* Vi1 = (const v2f*)(Vi + (size_t)(16 + row) * N_);
  const v2f* Ba  = (const v2f*)(Ab + (size_t)(n0 + row) * N_);
  v8f accR0 = {}, accI0 = {}, accR1 = {}, accI1 = {};
  #pragma unroll 4
  for (int k0 = 0; k0 < N_; k0 += 4) {
    int kk = (k0 + kh) >> 1;                   // b64-granular index (kb is even)
    v2f ba = Ba[kk];
    v2f r0 = Vr0[kk], i0 = Vi0[kk], r1 = Vr1[kk], i1 = Vi1[kk];
    accR0 = wmma4(r0, ba, accR0);
    accI0 = wmma4(i0, ba, accI0);
    accR1 = wmma4(r1, ba, accR1);
    accI1 = wmma4(i1, ba, accI1);
  }
  int colw = lane & 15;
  int moff = (lane >> 4) << 3;
  #pragma unroll
  for (int v = 0; v < 8; ++v) {
    size_t o0 = ((size_t)bh * C_ + v + moff) * N_ + n0 + colw;        // c rows 0..15
    size_t o1 = ((size_t)bh * C_ + 16 + v + moff) * N_ + n0 + colw;   // c rows 16..31
    Vor[o0] = accR0[v]; Voi[o0] = accI0[v];
    Vor[o1] = accR1[v]; Voi[o1] = accI1[v];
  }
}

// ---------------------------------------------------------------------
// K5: V-side positional term. Bucket-reduce T[part][r] = sum_m A*basis via
// LDS ds_add_f32 atomics, then Vo[b,h,c,n] += sum_r pw1[1,h,q,r]*T[part][r].
// ---------------------------------------------------------------------
__global__ void vo_pos(const float* __restrict__ A,
                       const float* __restrict__ pw0, const float* __restrict__ pw1,
                       const float* __restrict__ bas0, const float* __restrict__ bas1,
                       const int* __restrict__ radii,
                       float* __restrict__ Vor, float* __restrict__ Voi, int NR) {
  __shared__ float T[4][MAXR][2];
  int tid = threadIdx.x;
  int idx = blockIdx.x;
  int n  = idx & (N_ - 1);
  int bh = idx >> 9;
  int h  = bh & 3;
  for (int t = tid; t < 4 * MAXR * 2; t += 256) (&T[0][0][0])[t] = 0.f;
  __syncthreads();
  for (int m = tid; m < N_; m += 256) {
    float a = A[((size_t)bh * N_ + n) * N_ + m];
    int r = radii[n * N_ + m];
    float b0r = bas0[2 * ((size_t)n * N_ + m)];
    float b0i = bas0[2 * ((size_t)n * N_ + m) + 1];
    atomicAdd(&T[0][r][0], a * b0r); atomicAdd(&T[0][r][1], a * b0i);
    #pragma unroll
    for (int d = 0; d < 3; ++d) {
      float b1r = bas1[2 * (((size_t)n * 3 + d) * N_ + m)];
      float b1i = bas1[2 * (((size_t)n * 3 + d) * N_ + m) + 1];
      atomicAdd(&T[1 + d][r][0], a * b1r); atomicAdd(&T[1 + d][r][1], a * b1i);
    }
  }
  __syncthreads();
  if (tid < C_) {
    int c = tid;
    int part = (c < 8) ? 0 : 1 + ((c - 8) >> 3);
    int q    = (c < 8) ? c : ((c - 8) & 7);
    const float* pw = (part == 0) ? pw0 : pw1;
    float sr = 0.f, si = 0.f;
    for (int r = 0; r < NR; ++r) {
      size_t po = 2 * ((size_t)((1 * NH + h) * QD_ + q) * NR + r);   // pos component 1
      float pr = pw[po], pi = pw[po + 1];
      float tr = T[part][r][0], ti = T[part][r][1];
      sr += pr * tr - pi * ti;
      si += pr * ti + pi * tr;
    }
    size_t o = ((size_t)bh * C_ + c) * N_ + n;
    Vor[o] += sr; Voi[o] += si;
  }
}

// ---------------------------------------------------------------------
// K6: output projection, (q,h)->dim interleave, write complex64 interleaved.
// ---------------------------------------------------------------------
__global__ void out_proj(const float* __restrict__ Vor, const float* __restrict__ Voi,
                         const float* __restrict__ out0, const float* __restrict__ out1,
                         float* __restrict__ dst) {
  int gid = blockIdx.x * blockDim.x + threadIdx.x;   // 262144 threads
  int n = gid & (N_ - 1); int u = gid >> 9;
  int i = u & 31;          u >>= 5;
  int dl = u & 3;          int b = u >> 2;
  int l = (dl == 0) ? 0 : 1;
  int d = (dl == 0) ? 0 : dl - 1;
  const float* om = (l == 0) ? out0 : out1;
  int coff = (l == 0) ? 0 : 8 + d * QD_;
  float ar = 0.f, ai = 0.f;
  for (int j = 0; j < C_; ++j) {
    int hh = j & 3, q = j >> 2;                // dim j = q*nh + h (output fuse order)
    size_t vo = ((size_t)(b * NH + hh) * C_ + coff + q) * N_ + n;
    float vr = Vor[vo], vi = Voi[vo];
    float wr = om[2 * (i * C_ + j)], wi = om[2 * (i * C_ + j) + 1];
    ar += wr * vr - wi * vi;
    ai += wr * vi + wi * vr;
  }
  size_t cidx;
  if (l == 0) cidx = ((size_t)b * C_ + i) * N_ + n;
  else        cidx = (size_t)B_ * C_ * N_ + (((size_t)b * 3 + d) * C_ + i) * N_ + n;
  dst[2 * cidx]     = ar;
  dst[2 * cidx + 1] = ai;
}

// ---------------------------------------------------------------------
extern "C" void kernel_launch(void* const* d_in, const int* in_sizes, int n_in,
                              void* d_out, int out_size, void* d_ws, size_t ws_size,
                              hipStream_t stream) {
  const float* x0   = (const float*)d_in[0];
  const float* x1   = (const float*)d_in[1];
  const float* emb0 = (const float*)d_in[2];
  const float* emb1 = (const float*)d_in[3];
  const float* out0 = (const float*)d_in[4];
  const float* out1 = (const float*)d_in[5];
  const float* pw0  = (const float*)d_in[6];
  const float* pw1  = (const float*)d_in[7];
  const float* bas0 = (const float*)d_in[8];
  const float* bas1 = (const float*)d_in[9];
  const int*   rad  = (const int*)d_in[10];

  int NR = in_sizes[6] / (2 * NH * QD_);   // pos_w0 = [2,4,1,8,NR] complex elems
  if (NR > MAXR) NR = MAXR;

  // workspace layout (floats)
  float* ws   = (float*)d_ws;
  float* QKVr = ws;                         // 3*16*32*512   =   786432
  float* QKVi = QKVr + 786432;              //                   786432
  float* Ar   = QKVi + 786432;              // 16*512*512    =  4194304
  float* Ai   = Ar + 4194304;               //                  4194304
  float* Vor  = Ai + 4194304;               // 16*32*512     =   262144
  float* Voi  = Vor + 262144;               //                   262144
  (void)ws_size; (void)n_in; (void)out_size;

  qkv_proj    <<<3072, 256, 0, stream>>>(x0, x1, emb0, emb1, QKVr, QKVi);
  attn_scores <<<1024, 256, 0, stream>>>(QKVr, QKVi, Ar, Ai);
  bias_softmax<<<8192, 256, 0, stream>>>(QKVr, QKVi, pw0, pw1, bas0, bas1, rad, Ar, Ai, NR);
  vo_gemm     <<<  64, 256, 0, stream>>>(QKVr, QKVi, Ar, Vor, Voi);
  vo_pos      <<<8192, 256, 0, stream>>>(Ar, pw0, pw1, bas0, bas1, rad, Vor, Voi, NR);
  out_proj    <<<1024, 256, 0, stream>>>(Vor, Voi, out0, out1, (float*)d_out);
}